// Model_69131793597027
// MI455X (gfx1250) — compile-verified
//
#include <hip/hip_runtime.h>

#define FEAT 64
#define NHEAD 16
#define HD 4
#define S_LEN 4097          // SEQ + EOS
#define S_PAD 4112          // 257 * 16
#define NT 257              // 16-wide tiles covering S_PAD
#define FF_DIM 96
#define NLAB 1024
#define ID_EOS 100
#define ID_SOF 101
#define LN_EPS 1e-5f

typedef __attribute__((ext_vector_type(2))) float v2f;
typedef __attribute__((ext_vector_type(8))) float v8f;

__device__ __forceinline__ float sigmoidf_(float x) { return 1.0f / (1.0f + __expf(-x)); }

// ---------------------------------------------------------------------------
// Kernel A: embedding lookup + QKV projection into per-head [h][s][d] layout.
// One block per (padded) row, 64 threads; thread j computes q/k/v feature j.
// ---------------------------------------------------------------------------
__global__ __launch_bounds__(64) void k_embed_qkv(
    const int* __restrict__ inpts, const float* __restrict__ emb,
    const float* __restrict__ w, const float* __restrict__ b,
    float* __restrict__ X, float* __restrict__ Q,
    float* __restrict__ K, float* __restrict__ V)
{
    const int i = blockIdx.x;
    const int j = threadIdx.x;            // 0..63
    __shared__ float xrow[FEAT];

    const int h = j >> 2, d = j & 3;
    const size_t qi = (size_t)h * S_PAD * HD + (size_t)i * HD + d;

    if (i >= S_LEN) {                     // zero padding rows
        Q[qi] = 0.0f; K[qi] = 0.0f; V[qi] = 0.0f;
        X[(size_t)i * FEAT + j] = 0.0f;
        return;
    }
    const int tok = (i < S_LEN - 1) ? inpts[i] : ID_EOS;
    xrow[j] = emb[tok * FEAT + j];
    __syncthreads();

    float aq = b[j], ak = b[FEAT + j], av = b[2 * FEAT + j];
#pragma unroll 8
    for (int f = 0; f < FEAT; ++f) {
        const float xv = xrow[f];
        aq += w[j * FEAT + f] * xv;
        ak += w[(FEAT + j) * FEAT + f] * xv;
        av += w[(2 * FEAT + j) * FEAT + f] * xv;
    }
    X[(size_t)i * FEAT + j] = xrow[j];
    Q[qi] = aq; K[qi] = ak; V[qi] = av;
}

// ---------------------------------------------------------------------------
// Kernel B: two-pass flash attention. One wave (32 lanes) per (q-tile, head).
// Scores via V_WMMA_F32_16X16X4_F32 (16x16 tile = Q[16x4] * K^T[4x16]).
// Pass 1: per-lane row max (no cross-lane ops in loop).
// Pass 2: exp(s - m_row), per-lane l and P*V partials (K/V are L2-hot).
// Cross-lane butterflies (ds_bpermute) happen only once per wave, outside
// the K loops -- no s_wait_dscnt serialization in the hot path.
// ---------------------------------------------------------------------------
__global__ __launch_bounds__(32) void k_attn(
    const float* __restrict__ Q, const float* __restrict__ K,
    const float* __restrict__ V, float* __restrict__ O)
{
    const int qt = blockIdx.x;
    const int h  = blockIdx.y;
    const int lane = threadIdx.x;         // 0..31 (wave32)
    const int half = lane >> 4;           // K-dim pair selector
    const int n    = lane & 15;           // row (A) / column (B,D) index
    const int d0   = half * 2;
    const int qb   = qt * 16;

    const float* __restrict__ Qh = Q + (size_t)h * S_PAD * HD;
    const float* __restrict__ Kh = K + (size_t)h * S_PAD * HD;
    const float* __restrict__ Vh = V + (size_t)h * S_PAD * HD;

    // A matrix (16x4): lane<16 -> (K0,K1) of row qb+n; lane>=16 -> (K2,K3).
    // Fold 1/sqrt(HD)=0.5 scaling into Q.
    v2f a;
    a.x = 0.5f * Qh[(qb + n) * HD + d0];
    a.y = 0.5f * Qh[(qb + n) * HD + d0 + 1];

    // ---------------- Pass 1: per-lane row maxima ----------------
    float m[8];
#pragma unroll
    for (int r = 0; r < 8; ++r) m[r] = -1e30f;

    for (int kt = 0; kt < NT; ++kt) {
        const int kb = kt * 16;
        v2f bm;
        bm.x = Kh[(kb + n) * HD + d0];
        bm.y = Kh[(kb + n) * HD + d0 + 1];
        if (kt + 1 < NT)
            __builtin_prefetch(Kh + (size_t)(kb + 16 + n) * HD, 0, 1);

        v8f c = {0.f, 0.f, 0.f, 0.f, 0.f, 0.f, 0.f, 0.f};
        v8f s = __builtin_amdgcn_wmma_f32_16x16x4_f32(
            false, a, false, bm, (short)0, c, false, false);

        const bool kvalid = (kb + n) < S_LEN;
#pragma unroll
        for (int r = 0; r < 8; ++r) {
            const float sv = kvalid ? s[r] : -1e30f;
            m[r] = fmaxf(m[r], sv);
        }
    }
    // One-time butterfly max across the 16 lanes of each half.
#pragma unroll
    for (int r = 0; r < 8; ++r) {
        m[r] = fmaxf(m[r], __shfl_xor(m[r], 1, 32));
        m[r] = fmaxf(m[r], __shfl_xor(m[r], 2, 32));
        m[r] = fmaxf(m[r], __shfl_xor(m[r], 4, 32));
        m[r] = fmaxf(m[r], __shfl_xor(m[r], 8, 32));
    }

    // ---------------- Pass 2: exp + accumulate ----------------
    float l[8], oacc[8][4];
#pragma unroll
    for (int r = 0; r < 8; ++r) {
        l[r] = 0.0f;
        oacc[r][0] = oacc[r][1] = oacc[r][2] = oacc[r][3] = 0.0f;
    }

    for (int kt = 0; kt < NT; ++kt) {
        const int kb = kt * 16;
        v2f bm;
        bm.x = Kh[(kb + n) * HD + d0];
        bm.y = Kh[(kb + n) * HD + d0 + 1];
        const float4 vr = *(const float4*)(Vh + (size_t)(kb + n) * HD);
        if (kt + 1 < NT)
            __builtin_prefetch(Vh + (size_t)(kb + 16 + n) * HD, 0, 1);

        v8f c = {0.f, 0.f, 0.f, 0.f, 0.f, 0.f, 0.f, 0.f};
        v8f s = __builtin_amdgcn_wmma_f32_16x16x4_f32(
            false, a, false, bm, (short)0, c, false, false);

        const bool kvalid = (kb + n) < S_LEN;
#pragma unroll
        for (int r = 0; r < 8; ++r) {
            const float sv = kvalid ? s[r] : -1e30f;
            const float p  = __expf(sv - m[r]);   // 0 for masked columns
            l[r] += p;
            oacc[r][0] += p * vr.x;
            oacc[r][1] += p * vr.y;
            oacc[r][2] += p * vr.z;
            oacc[r][3] += p * vr.w;
        }
    }

    // One-time butterfly sums across the 16 lanes of each half.
#pragma unroll
    for (int r = 0; r < 8; ++r) {
        l[r] += __shfl_xor(l[r], 1, 32);
        l[r] += __shfl_xor(l[r], 2, 32);
        l[r] += __shfl_xor(l[r], 4, 32);
        l[r] += __shfl_xor(l[r], 8, 32);
#pragma unroll
        for (int d = 0; d < 4; ++d) {
            float t = oacc[r][d];
            t += __shfl_xor(t, 1, 32);
            t += __shfl_xor(t, 2, 32);
            t += __shfl_xor(t, 4, 32);
            t += __shfl_xor(t, 8, 32);
            oacc[r][d] = t;
        }
    }

    // Store: half 0 -> rows qb+0..7, half 1 -> rows qb+8..15.
#pragma unroll
    for (int r = 0; r < 8; ++r) {
        const int row = qb + half * 8 + r;
        if (row < S_LEN) {
            const float invl = 1.0f / l[r];
            if (n == 2 * r) {
                v2f o2 = { oacc[r][0] * invl, oacc[r][1] * invl };
                *(v2f*)(O + (size_t)row * FEAT + h * HD) = o2;
            }
            if (n == 2 * r + 1) {
                v2f o2 = { oacc[r][2] * invl, oacc[r][3] * invl };
                *(v2f*)(O + (size_t)row * FEAT + h * HD + 2) = o2;
            }
        }
    }
}

// ---------------------------------------------------------------------------
// Kernel C: out-proj + residual + LN1 + FF(relu) + residual + LN2 -> embs.
// One block per row, 96 threads (3 waves).
// ---------------------------------------------------------------------------
__global__ __launch_bounds__(96) void k_encoder_tail(
    const float* __restrict__ X, const float* __restrict__ O,
    const float* __restrict__ wo, const float* __restrict__ bo,
    const float* __restrict__ g1, const float* __restrict__ b1,
    const float* __restrict__ w1, const float* __restrict__ fb1,
    const float* __restrict__ w2, const float* __restrict__ fb2,
    const float* __restrict__ g2, const float* __restrict__ b2,
    float* __restrict__ E)
{
    const int i = blockIdx.x;
    const int t = threadIdx.x;
    __shared__ float orow[FEAT], xrow[FEAT], tbuf[FEAT], x1b[FEAT], hbuf[FF_DIM];
    __shared__ float s_mu, s_rsig;

    if (t < FEAT) {
        orow[t] = O[(size_t)i * FEAT + t];
        xrow[t] = X[(size_t)i * FEAT + t];
    }
    __syncthreads();

    if (t < FEAT) {
        float acc = bo[t];
#pragma unroll 8
        for (int f = 0; f < FEAT; ++f) acc += wo[t * FEAT + f] * orow[f];
        tbuf[t] = xrow[t] + acc;
    }
    __syncthreads();
    if (t == 0) {
        float su = 0.f;
        for (int f = 0; f < FEAT; ++f) su += tbuf[f];
        const float mu = su / FEAT;
        float vv = 0.f;
        for (int f = 0; f < FEAT; ++f) { float d = tbuf[f] - mu; vv += d * d; }
        s_mu = mu; s_rsig = rsqrtf(vv / FEAT + LN_EPS);
    }
    __syncthreads();
    if (t < FEAT) x1b[t] = (tbuf[t] - s_mu) * s_rsig * g1[t] + b1[t];
    __syncthreads();

    if (t < FF_DIM) {
        float acc = fb1[t];
#pragma unroll 8
        for (int f = 0; f < FEAT; ++f) acc += w1[t * FEAT + f] * x1b[f];
        hbuf[t] = fmaxf(acc, 0.0f);
    }
    __syncthreads();
    if (t < FEAT) {
        float acc = fb2[t];
#pragma unroll 8
        for (int f = 0; f < FF_DIM; ++f) acc += w2[t * FF_DIM + f] * hbuf[f];
        tbuf[t] = x1b[t] + acc;
    }
    __syncthreads();
    if (t == 0) {
        float su = 0.f;
        for (int f = 0; f < FEAT; ++f) su += tbuf[f];
        const float mu = su / FEAT;
        float vv = 0.f;
        for (int f = 0; f < FEAT; ++f) { float d = tbuf[f] - mu; vv += d * d; }
        s_mu = mu; s_rsig = rsqrtf(vv / FEAT + LN_EPS);
    }
    __syncthreads();
    if (t < FEAT) E[(size_t)i * FEAT + t] = (tbuf[t] - s_mu) * s_rsig * g2[t] + b2[t];
}

// ---------------------------------------------------------------------------
// Kernel D: sequential pointer decoder. Single 1024-thread block, 1024 steps.
// Per step: GRU (h0=0) -> hq, logits = embs @ hq, fused logsumexp + argmax
// via wave32 shfl reduction + LDS combine, greedy feedback.
// ---------------------------------------------------------------------------
__global__ __launch_bounds__(1024) void k_decoder(
    const float* __restrict__ E, const float* __restrict__ emb,
    const int* __restrict__ labels,
    const float* __restrict__ wih, const float* __restrict__ bih,
    const float* __restrict__ bhh,
    float* __restrict__ out)
{
    const int t = threadIdx.x;
    __shared__ float dec_in[FEAT], gi[3 * FEAT], hq[FEAT];
    __shared__ float red_m[32], red_s[32], red_bv[32];
    __shared__ int   red_bi[32];
    __shared__ float s_label_logit;
    __shared__ int   s_best;

    if (t < FEAT) dec_in[t] = emb[ID_SOF * FEAT + t];
    float lossSum = 0.0f;

    for (int step = 0; step < NLAB; ++step) {
        const int label = labels[step];
        __syncthreads();                          // dec_in ready

        if (t < 3 * FEAT) {
            float acc = bih[t];
#pragma unroll 8
            for (int f = 0; f < FEAT; ++f) acc += wih[t * FEAT + f] * dec_in[f];
            gi[t] = acc;
        }
        __syncthreads();
        if (t < FEAT) {
            const float r = sigmoidf_(gi[t] + bhh[t]);
            const float z = sigmoidf_(gi[FEAT + t] + bhh[FEAT + t]);
            const float nn = tanhf(gi[2 * FEAT + t] + r * bhh[2 * FEAT + t]);
            hq[t] = (1.0f - z) * nn;              // h0 = 0
        }
        __syncthreads();

        // logits over 4097 rows, strided across 1024 threads
        float m = -1e30f, s = 0.0f, bv = -1e30f;
        int bi = 0x7fffffff;
        for (int j = t; j < S_LEN; j += 1024) {
            const float* __restrict__ er = E + (size_t)j * FEAT;
            float lg = 0.0f;
#pragma unroll 8
            for (int f = 0; f < FEAT; ++f) lg += er[f] * hq[f];
            if (lg > bv) { bv = lg; bi = j; }     // first-max semantics
            const float mn = fmaxf(m, lg);
            s = s * __expf(m - mn) + __expf(lg - mn);
            m = mn;
            if (j == label) s_label_logit = lg;
        }
        // wave32 butterfly reduce
#pragma unroll
        for (int off = 1; off < 32; off <<= 1) {
            const float m2  = __shfl_xor(m, off, 32);
            const float s2  = __shfl_xor(s, off, 32);
            const float bv2 = __shfl_xor(bv, off, 32);
            const int   bi2 = __shfl_xor(bi, off, 32);
            const float mn = fmaxf(m, m2);
            s = s * __expf(m - mn) + s2 * __expf(m2 - mn);
            m = mn;
            if (bv2 > bv || (bv2 == bv && bi2 < bi)) { bv = bv2; bi = bi2; }
        }
        if ((t & 31) == 0) {
            const int w = t >> 5;
            red_m[w] = m; red_s[w] = s; red_bv[w] = bv; red_bi[w] = bi;
        }
        __syncthreads();
        if (t == 0) {
            float M = red_m[0], Ss = red_s[0], BV = red_bv[0];
            int BI = red_bi[0];
            for (int w = 1; w < 32; ++w) {
                const float m2 = red_m[w], s2 = red_s[w];
                const float mn = fmaxf(M, m2);
                Ss = Ss * __expf(M - mn) + s2 * __expf(m2 - mn);
                M = mn;
                const float bv2 = red_bv[w]; const int bi2 = red_bi[w];
                if (bv2 > BV || (bv2 == BV && bi2 < BI)) { BV = bv2; BI = bi2; }
            }
            lossSum += M + __logf(Ss) - s_label_logit;
            s_best = BI;
        }
        __syncthreads();
        if (t < FEAT) dec_in[t] = E[(size_t)s_best * FEAT + t];
    }
    if (t == 0) out[0] = lossSum;
}

// ---------------------------------------------------------------------------
extern "C" void kernel_launch(void* const* d_in, const int* in_sizes, int n_in,
                              void* d_out, int out_size, void* d_ws, size_t ws_size,
                              hipStream_t stream)
{
    (void)in_sizes; (void)n_in; (void)out_size; (void)ws_size;

    const int*   inpts      = (const int*)  d_in[0];
    const int*   labels     = (const int*)  d_in[1];
    const float* emb_table  = (const float*)d_in[2];
    const float* attn_in_w  = (const float*)d_in[3];
    const float* attn_in_b  = (const float*)d_in[4];
    const float* attn_out_w = (const float*)d_in[5];
    const float* attn_out_b = (const float*)d_in[6];
    const float* ln1_g      = (const float*)d_in[7];
    const float* ln1_b      = (const float*)d_in[8];
    const float* ff1_w      = (const float*)d_in[9];
    const float* ff1_b      = (const float*)d_in[10];
    const float* ff2_w      = (const float*)d_in[11];
    const float* ff2_b      = (const float*)d_in[12];
    const float* ln2_g      = (const float*)d_in[13];
    const float* ln2_b      = (const float*)d_in[14];
    const float* gru_wih    = (const float*)d_in[15];
    /* gru_whh (d_in[16]) unused: h0 == 0 so whh @ h0 == 0 */
    const float* gru_bih    = (const float*)d_in[17];
    const float* gru_bhh    = (const float*)d_in[18];

    float* ws = (float*)d_ws;
    const size_t blk = (size_t)S_PAD * FEAT;   // 263168 floats per buffer
    float* X = ws;
    float* Q = X + blk;
    float* K = Q + blk;
    float* V = K + blk;
    float* O = V + blk;
    float* E = O + blk;

    k_embed_qkv<<<S_PAD, 64, 0, stream>>>(inpts, emb_table, attn_in_w, attn_in_b,
                                          X, Q, K, V);
    k_attn<<<dim3(NT, NHEAD), 32, 0, stream>>>(Q, K, V, O);
    k_encoder_tail<<<S_LEN, 96, 0, stream>>>(X, O, attn_out_w, attn_out_b,
                                             ln1_g, ln1_b, ff1_w, ff1_b,
                                             ff2_w, ff2_b, ln2_g, ln2_b, E);
    k_decoder<<<1, 1024, 0, stream>>>(E, emb_table, labels,
                                      gru_wih, gru_bih, gru_bhh, (float*)d_out);
}